// SymmetryAwareLossLoop_87780541596355
// MI455X (gfx1250) — compile-verified
//
#include <hip/hip_runtime.h>
#include <math.h>

// ---------------------------------------------------------------------------
// Symmetry-aware rotation loss for P6122 (point group 622), B=16384, C=32, S=12
//
// tr[b,c,s] = <R_pred[b,c], rot[s] @ R_gt[b]>_F  is a per-batch GEMM
//   (M=32 candidates) x (K=9 elems, padded to 12) x (N=12 syms, padded to 16)
// executed with V_WMMA_F32_16X16X4_F32 (2 M-tiles x 3 K-steps = 6 WMMA/batch).
// min theta = acos(clip(max tr)) by monotonicity -> one acos per batch.
// ---------------------------------------------------------------------------

typedef float v2f __attribute__((ext_vector_type(2)));
typedef float v8f __attribute__((ext_vector_type(8)));

constexpr int   kB          = 16384;
constexpr int   kC          = 32;
constexpr int   kS          = 12;
constexpr float kEPS        = 1e-7f;
constexpr int   kThreads    = 256;                 // 8 waves of 32
constexpr int   kWavesPerBlk= kThreads / 32;
constexpr int   kNumBlocks  = 1024;                // 8192 waves -> 2 batches/wave
constexpr int   kTotalWaves = kNumBlocks * kWavesPerBlk;

__global__ __launch_bounds__(kThreads)
void sym_loss_main(const float* __restrict__ R_pred,   // (B, C, 3, 3)
                   const float* __restrict__ R_gt,     // (B, 3, 3)
                   const float* __restrict__ rot_mats, // (S, 3, 3)
                   float* __restrict__ partial)        // (kNumBlocks,)
{
    __shared__ float s_rot[kS * 9];          // 432 B, block-resident constants
    __shared__ float s_wavesum[kWavesPerBlk];

    const int tid = threadIdx.x;
    if (tid < kS * 9) s_rot[tid] = rot_mats[tid];
    __syncthreads();

    const int  lane  = tid & 31;
    const int  wave  = tid >> 5;
    const int  gwave = blockIdx.x * kWavesPerBlk + wave;
    const int  h     = lane >> 4;            // half-wave select (K striping)
    const int  s     = lane & 15;            // symmetry column N = s
    const bool svalid = (s < kS);

    float wsum = 0.0f;

    for (int b = gwave; b < kB; b += kTotalWaves) {
        // ---- R_gt[b] : wave-uniform 9 floats -----------------------------
        const float* gq = R_gt + (long)b * 9;
        float g[9];
#pragma unroll
        for (int j = 0; j < 9; ++j) g[j] = gq[j];

        // ---- B operand: Bsym[k][s] = sum_m rot[s,i,m] * R_gt[m,j], k=3i+j
        // Lane owns k = 4t + 2h + d  (t=0..2, d=0..1); k>=9 is zero padding.
        float bb[6];
#pragma unroll
        for (int t = 0; t < 3; ++t) {
#pragma unroll
            for (int d = 0; d < 2; ++d) {
                const int k = 4 * t + 2 * h + d;
                float v = 0.0f;
                if (svalid && k < 9) {
                    const int i = k / 3;
                    const int j = k - 3 * i;
                    const float* r = &s_rot[s * 9 + i * 3];
                    v = fmaf(r[0], g[j], fmaf(r[1], g[3 + j], r[2] * g[6 + j]));
                }
                bb[2 * t + d] = v;
            }
        }

        // ---- A operand: R_pred[b, c, k] with c = s + 16*tile --------------
        // Across the (L, L+16) lane pair each candidate row's 9 floats are
        // read exactly once: no redundant HBM traffic.
        const float* pbase = R_pred + ((long)b * kC) * 9;
        float aa[2][6];
#pragma unroll
        for (int tile = 0; tile < 2; ++tile) {
            const float* pr = pbase + (s + 16 * tile) * 9;
#pragma unroll
            for (int t = 0; t < 3; ++t) {
#pragma unroll
                for (int d = 0; d < 2; ++d) {
                    const int k = 4 * t + 2 * h + d;
                    aa[tile][2 * t + d] = (k < 9) ? pr[k] : 0.0f;
                }
            }
        }

        // ---- 6x V_WMMA_F32_16X16X4_F32 ------------------------------------
        v8f acc0 = {};   // candidates  0..15
        v8f acc1 = {};   // candidates 16..31
#pragma unroll
        for (int t = 0; t < 3; ++t) {
            v2f a0; a0.x = aa[0][2 * t]; a0.y = aa[0][2 * t + 1];
            v2f a1; a1.x = aa[1][2 * t]; a1.y = aa[1][2 * t + 1];
            v2f bv; bv.x = bb[2 * t];    bv.y = bb[2 * t + 1];
            acc0 = __builtin_amdgcn_wmma_f32_16x16x4_f32(
                       false, a0, false, bv, (short)0, acc0, false, false);
            acc1 = __builtin_amdgcn_wmma_f32_16x16x4_f32(
                       false, a1, false, bv, (short)0, acc1, false, false);
        }

        // ---- max over (c,s): per-lane over 16 VGPRs, then cross-lane ------
        float m = -3.0e38f;
#pragma unroll
        for (int r = 0; r < 8; ++r) {
            m = fmaxf(m, acc0[r]);
            m = fmaxf(m, acc1[r]);
        }
        if (!svalid) m = -3.0e38f;           // mask padding columns N=12..15
#pragma unroll
        for (int off = 16; off >= 1; off >>= 1)
            m = fmaxf(m, __shfl_xor(m, off, 32));

        // min theta = acos(clip((max tr - 1)/2))  (acos monotone decreasing)
        const float cosv = fminf(fmaxf((m - 1.0f) * 0.5f, -1.0f + kEPS),
                                 1.0f - kEPS);
        wsum += acosf(cosv);                 // identical across the wave
    }

    if (lane == 0) s_wavesum[wave] = wsum;
    __syncthreads();
    if (tid == 0) {
        float t = 0.0f;
#pragma unroll
        for (int i = 0; i < kWavesPerBlk; ++i) t += s_wavesum[i];
        partial[blockIdx.x] = t;
    }
}

__global__ __launch_bounds__(kThreads)
void sym_loss_reduce(const float* __restrict__ partial, float* __restrict__ out)
{
    __shared__ float sm[kThreads];
    const int tid = threadIdx.x;
    float t = 0.0f;
    for (int i = tid; i < kNumBlocks; i += kThreads) t += partial[i];
    sm[tid] = t;
    __syncthreads();
#pragma unroll
    for (int s2 = kThreads / 2; s2 > 0; s2 >>= 1) {
        if (tid < s2) sm[tid] += sm[tid + s2];
        __syncthreads();
    }
    if (tid == 0) out[0] = sm[0] * (1.0f / (float)kB);
}

extern "C" void kernel_launch(void* const* d_in, const int* in_sizes, int n_in,
                              void* d_out, int out_size, void* d_ws, size_t ws_size,
                              hipStream_t stream)
{
    (void)in_sizes; (void)n_in; (void)out_size; (void)ws_size;
    const float* R_pred   = (const float*)d_in[0];
    const float* R_gt     = (const float*)d_in[1];
    const float* rot_mats = (const float*)d_in[2];
    float*       out      = (float*)d_out;
    float*       partial  = (float*)d_ws;    // kNumBlocks floats (4 KB)

    sym_loss_main<<<kNumBlocks, kThreads, 0, stream>>>(R_pred, R_gt, rot_mats,
                                                       partial);
    sym_loss_reduce<<<1, kThreads, 0, stream>>>(partial, out);
}